// FDGN_58506044506617
// MI455X (gfx1250) — compile-verified
//
#include <hip/hip_runtime.h>
#include <math.h>

#define NFEAT 256
#define NHID  128
#define NCLS  16

typedef __attribute__((ext_vector_type(2))) float        v2f;
typedef __attribute__((ext_vector_type(8))) float        v8f;
typedef __attribute__((ext_vector_type(4))) unsigned int v4u;
typedef __attribute__((ext_vector_type(4))) int          v4i;
typedef __attribute__((ext_vector_type(8))) int          v8i;

// ---------------- degree / normalization ----------------

__global__ __launch_bounds__(256) void gcn_zero_k(float* __restrict__ colsum,
                                                  float* __restrict__ ncr_acc, int N) {
  int i = blockIdx.x * 256 + threadIdx.x;
  if (i < N) colsum[i] = 0.0f;
  if (i == 0) *ncr_acc = 0.0f;
}

__global__ __launch_bounds__(256) void gcn_colsum_k(const int* __restrict__ col,
                                                    const float* __restrict__ w,
                                                    float* __restrict__ colsum, int E) {
  int e = blockIdx.x * 256 + threadIdx.x;
  if (e < E) atomicAdd(&colsum[col[e]], w[e]);
}

__global__ __launch_bounds__(256) void gcn_norm_k(const float* __restrict__ colsum,
                                                  float* __restrict__ dinv,
                                                  float* __restrict__ rinv, int N) {
  int i = blockIdx.x * 256 + threadIdx.x;
  if (i >= N) return;
  float es  = colsum[i];
  float deg = es + 1.0f;                 // self-loop weight 1.0; deg >= 1 always
  dinv[i] = rsqrtf(deg);
  rinv[i] = (es > 0.0f) ? (1.0f / es) : 0.0f;
}

// ---------------- TDM 2D tile load: global fp32 [tile_h x width] -> LDS ------
// D# per CDNA5 ISA ch.8: group0 {count, lds_addr, global_addr, type=2},
// group1 {data_size=4B, tensor_dim0=width, tensor_dim1=tile_h, tile_dim0=width,
//         tile_dim1=tile_h, tensor_dim0_stride=width}. 2D => groups 2/3 NULL/0.

#if __has_builtin(__builtin_amdgcn_tensor_load_to_lds)
#define GCN_HAS_TDM 1
__device__ __forceinline__ void tdm_load_tile_f32(unsigned lds_off_bytes,
                                                  const float* gptr,
                                                  unsigned width,     // elements per row
                                                  unsigned tile_h) {  // rows
  unsigned long long ga = (unsigned long long)(uintptr_t)gptr;
  v4u g0 = { 1u,                                   // count=1 (valid descriptor)
             lds_off_bytes,                        // lds_addr
             (unsigned)ga,                         // global_addr[31:0]
             (unsigned)(ga >> 32) | (2u << 30) };  // global_addr[56:32] | type=2
  v8i g1 = { (int)(2u << 16),                                          // data_size=4B
             (int)((width & 0xFFFFu) << 16),                           // tensor_dim0 lo16
             (int)(((tile_h & 0xFFFFu) << 16) | ((width >> 16) & 0xFFFFu)), // dim1 lo | dim0 hi
             (int)(((width & 0xFFFFu) << 16) | ((tile_h >> 16) & 0xFFFFu)), // tile_dim0 | dim1 hi
             (int)(tile_h & 0xFFFFu),                                  // tile_dim1 (tile_dim2=0)
             (int)width,                                               // tensor_dim0_stride lo32
             0, 0 };
  v4i gz = { 0, 0, 0, 0 };
#if defined(__clang_major__) && (__clang_major__ >= 23)
  v8i gz8 = { 0, 0, 0, 0, 0, 0, 0, 0 };
  __builtin_amdgcn_tensor_load_to_lds(g0, g1, gz, gz, gz8, 0);
#else
  __builtin_amdgcn_tensor_load_to_lds(g0, g1, gz, gz, 0);
#endif
  __builtin_amdgcn_s_wait_tensorcnt(0);
}
#else
#define GCN_HAS_TDM 0
#endif

// ---------------- fp32 WMMA GEMM with LDS-staged A panel ----------------
// Block = 256 threads (8 waves). Block covers (8/tilesN) M-tiles x NC cols.
// A panel [rowsPerBlk x K] is DMA'd to LDS by the TDM (wave 0), shared by all
// waves. B ([K x NC], small, L2-resident) is read directly from global.
// V_WMMA_F32_16X16X4_F32: A frag lane(g,m) = A[m][k+2g+{0,1}],
//                         B frag lane(g,m) = B[k+2g+{0,1}][n+m],
//                         D component v -> row v+8g, col m.

template <int K, int NC, bool RELU>
__global__ __launch_bounds__(256) void gcn_gemm_wmma_k(const float* __restrict__ A,
                                                       const float* __restrict__ B,
                                                       float* __restrict__ C, int M) {
  constexpr int tilesN      = NC / 16;
  constexpr int mtilesPerBlk = 8 / tilesN;
  constexpr int rowsPerBlk   = 16 * mtilesPerBlk;
  __shared__ float lds_a[rowsPerBlk * K];

  const int mbase = blockIdx.x * rowsPerBlk;
  const int rows  = (M - mbase < rowsPerBlk) ? (M - mbase) : rowsPerBlk;

#if GCN_HAS_TDM
  if (threadIdx.x < 32) {  // one wave issues the DMA and waits TENSORcnt->0
    tdm_load_tile_f32((unsigned)(uintptr_t)(void*)&lds_a[0],
                      A + (size_t)mbase * K, (unsigned)K, (unsigned)rows);
  }
#else
  for (int i = threadIdx.x; i < rows * K; i += 256)
    lds_a[i] = A[(size_t)mbase * K + i];
#endif
  __syncthreads();

  const int wave = threadIdx.x >> 5;
  const int lane = threadIdx.x & 31;
  const int mt   = wave / tilesN;         // M-tile within block
  const int tn   = wave % tilesN;         // N-tile
  const int g    = lane >> 4;
  const int m    = lane & 15;

  if (mt * 16 < rows) {                   // wave-uniform: EXEC all-ones for WMMA
    const float* arow = &lds_a[(mt * 16 + m) * K];
    const float* bcol = B + tn * 16 + m;

    v8f acc = {};
    for (int k = 0; k < K; k += 4) {
      v2f a, b;
      float a0 = arow[k + 2 * g];
      float a1 = arow[k + 2 * g + 1];
      if (RELU) { a0 = fmaxf(a0, 0.0f); a1 = fmaxf(a1, 0.0f); }
      a.x = a0; a.y = a1;
      b.x = bcol[(size_t)(k + 2 * g) * NC];
      b.y = bcol[(size_t)(k + 2 * g + 1) * NC];
      acc = __builtin_amdgcn_wmma_f32_16x16x4_f32(false, a, false, b,
                                                  (short)0, acc, false, false);
    }

    float* crow = C + (size_t)(mbase + mt * 16 + 8 * g) * NC + tn * 16 + m;
#pragma unroll
    for (int v = 0; v < 8; ++v) crow[(size_t)v * NC] = acc[v];
  }
}

// ---------------- self-loop + bias init: out = dinv^2 * hw + b ----------------

template <int F>
__global__ __launch_bounds__(256) void gcn_selfbias_k(const float* __restrict__ hw,
                                                      const float* __restrict__ dinv,
                                                      const float* __restrict__ bias,
                                                      float* __restrict__ out, int N) {
  int idx = blockIdx.x * 256 + threadIdx.x;
  if (idx >= N * F) return;
  int i = idx / F;
  int f = idx - i * F;
  float d = dinv[i];
  out[idx] = d * d * hw[idx] + bias[f];
}

// ---------------- edge scatter, F=128: one wave per edge, float4 per lane ----

__global__ __launch_bounds__(256) void gcn_scatter128_k(const int* __restrict__ row,
                                                        const int* __restrict__ col,
                                                        const float* __restrict__ w,
                                                        const float* __restrict__ dinv,
                                                        const float* __restrict__ hw,
                                                        float* __restrict__ out, int E) {
  int tid  = blockIdx.x * 256 + threadIdx.x;
  int e    = tid >> 5;
  int lane = tid & 31;
  if (e >= E) return;
  int r = row[e], c = col[e];
  float nrm = dinv[r] * w[e] * dinv[c];
  const float4 hv = *(const float4*)(hw + (size_t)r * NHID + lane * 4);
  float* o = out + (size_t)c * NHID + lane * 4;
  atomicAdd(o + 0, nrm * hv.x);
  atomicAdd(o + 1, nrm * hv.y);
  atomicAdd(o + 2, nrm * hv.z);
  atomicAdd(o + 3, nrm * hv.w);
}

// ---------------- edge scatter, F=16: 16 threads per edge ----------------

__global__ __launch_bounds__(256) void gcn_scatter16_k(const int* __restrict__ row,
                                                       const int* __restrict__ col,
                                                       const float* __restrict__ w,
                                                       const float* __restrict__ dinv,
                                                       const float* __restrict__ hw,
                                                       float* __restrict__ out, int E) {
  int tid = blockIdx.x * 256 + threadIdx.x;
  int e = tid >> 4;
  int f = tid & 15;
  if (e >= E) return;
  int r = row[e], c = col[e];
  float nrm = dinv[r] * w[e] * dinv[c];
  atomicAdd(out + (size_t)c * NCLS + f, nrm * hw[(size_t)r * NCLS + f]);
}

// ---------------- neighbor consistency (per-edge KL, 16 classes) ----------------

__global__ __launch_bounds__(256) void gcn_ncr_k(const int* __restrict__ row,
                                                 const int* __restrict__ col,
                                                 const float* __restrict__ w,
                                                 const float* __restrict__ rinv,
                                                 const float* __restrict__ y,
                                                 float* __restrict__ acc,
                                                 int E, float invN) {
  int e = blockIdx.x * 256 + threadIdx.x;
  if (e >= E) return;
  int r = row[e], c = col[e];
  const float* yr = y + (size_t)r * NCLS;   // target logits (detached)
  const float* yc = y + (size_t)c * NCLS;   // prediction logits
  float t[NCLS], p[NCLS];
  float mt = -__builtin_inff(), mp = -__builtin_inff();
#pragma unroll
  for (int k = 0; k < NCLS; ++k) {
    t[k] = yr[k]; p[k] = yc[k];
    mt = fmaxf(mt, t[k]); mp = fmaxf(mp, p[k]);
  }
  float st = 0.0f, sp = 0.0f;
#pragma unroll
  for (int k = 0; k < NCLS; ++k) {
    st += __expf(t[k] - mt);
    sp += __expf(p[k] - mp);
  }
  float lzt = mt + __logf(st);
  float lzp = mp + __logf(sp);
  float kl = 0.0f;
#pragma unroll
  for (int k = 0; k < NCLS; ++k) {
    float tp = __expf(t[k] - lzt);
    kl += tp * ((t[k] - lzt) - (p[k] - lzp));
  }
  atomicAdd(acc, w[e] * rinv[c] * kl * invN);
}

// ---------------- launch ----------------

static inline int cdiv_i(long long a, long long b) { return (int)((a + b - 1) / b); }

extern "C" void kernel_launch(void* const* d_in, const int* in_sizes, int n_in,
                              void* d_out, int out_size, void* d_ws, size_t ws_size,
                              hipStream_t stream) {
  const float* x   = (const float*)d_in[0];
  const int*   ei  = (const int*)  d_in[1];
  const float* w   = (const float*)d_in[2];
  const float* W1  = (const float*)d_in[3];
  const float* b1  = (const float*)d_in[4];
  const float* W2  = (const float*)d_in[5];
  const float* b2  = (const float*)d_in[6];

  const int N = in_sizes[0] / NFEAT;     // 50000
  const int E = in_sizes[2];             // 1600000
  const int* row = ei;
  const int* col = ei + E;

  float* ws     = (float*)d_ws;
  float* colsum = ws;
  float* dinv   = colsum + N;
  float* rinv   = dinv + N;
  float* hw1    = rinv + N;                      // [N, 128]
  float* out1   = hw1 + (size_t)N * NHID;        // [N, 128] (pre-ReLU hidden)
  float* hw2    = out1 + (size_t)N * NHID;       // [N, 16]

  float* logits  = (float*)d_out;                // [N, 16]
  float* ncr_acc = logits + (size_t)N * NCLS;    // scalar

  // 1) normalization terms
  gcn_zero_k  <<<cdiv_i(N, 256), 256, 0, stream>>>(colsum, ncr_acc, N);
  gcn_colsum_k<<<cdiv_i(E, 256), 256, 0, stream>>>(col, w, colsum, E);
  gcn_norm_k  <<<cdiv_i(N, 256), 256, 0, stream>>>(colsum, dinv, rinv, N);

  // 2) layer 1: hw1 = x @ W1 (fp32 WMMA, TDM-staged A), aggregate into out1
  {
    int blocks = cdiv_i(N, 16);                  // rowsPerBlk = 16 (8 N-tiles/block)
    gcn_gemm_wmma_k<NFEAT, NHID, false><<<blocks, 256, 0, stream>>>(x, W1, hw1, N);
  }
  gcn_selfbias_k<NHID><<<cdiv_i((long long)N * NHID, 256), 256, 0, stream>>>(hw1, dinv, b1, out1, N);
  gcn_scatter128_k<<<cdiv_i((long long)E * 32, 256), 256, 0, stream>>>(row, col, w, dinv, hw1, out1, E);

  // 3) layer 2: hw2 = relu(out1) @ W2 (ReLU fused), aggregate into logits
  {
    int blocks = cdiv_i(N, 128);                 // rowsPerBlk = 128 (1 N-tile, 8 M-tiles)
    gcn_gemm_wmma_k<NHID, NCLS, true><<<blocks, 256, 0, stream>>>(out1, W2, hw2, N);
  }
  gcn_selfbias_k<NCLS><<<cdiv_i((long long)N * NCLS, 256), 256, 0, stream>>>(hw2, dinv, b2, logits, N);
  gcn_scatter16_k<<<cdiv_i((long long)E * 16, 256), 256, 0, stream>>>(row, col, w, dinv, hw2, logits, E);

  // 4) neighbor-consistency scalar
  gcn_ncr_k<<<cdiv_i(E, 256), 256, 0, stream>>>(row, col, w, rinv, logits, ncr_acc, E, 1.0f / (float)N);
}